// QuantumMambaSSMCoreAdvanced_83863531421732
// MI455X (gfx1250) — compile-verified
//
#include <hip/hip_runtime.h>
#include <hip/hip_bf16.h>
#include <math.h>

// ---------------- problem constants ----------------
#define NQ      6
#define DIM     64          // 2^NQ amplitudes
#define BATCH   256
#define SEQ     512
#define OUTD    18          // 3*NQ
#define DTRANK  3
#define BT      16          // batch tile per block
#define NBLK    (BATCH / BT)
#define PI_F    3.14159265358979323846f

typedef __attribute__((ext_vector_type(2))) float v2f;
typedef __attribute__((ext_vector_type(8))) float v8f;

// =====================================================================
// Kernel A: per-(b,t) input projections:
//   x_dbc = angles @ Wx.T ; dt_angles = tanh(softplus(dt_raw@Wdt.T+bdt))*pi
//   C     = x_dbc[..., DTRANK+NQ:]
// =====================================================================
__global__ void qm_precompute_kernel(const float* __restrict__ ang,
                                     const float* __restrict__ Wx,
                                     const float* __restrict__ Wdt,
                                     const float* __restrict__ bdt,
                                     float* __restrict__ dtang,
                                     float* __restrict__ Cbuf) {
    int idx = blockIdx.x * blockDim.x + threadIdx.x;   // over BATCH*SEQ
    if (idx >= BATCH * SEQ) return;
    float a[NQ];
#pragma unroll
    for (int i = 0; i < NQ; ++i) a[i] = ang[(size_t)idx * NQ + i];

    float dtr[DTRANK];
#pragma unroll
    for (int j = 0; j < DTRANK; ++j) {
        float s = 0.f;
#pragma unroll
        for (int i = 0; i < NQ; ++i) s += a[i] * Wx[j * NQ + i];
        dtr[j] = s;
    }
#pragma unroll
    for (int j = 0; j < NQ; ++j) {              // C rows start at DTRANK+NQ = 9
        float s = 0.f;
#pragma unroll
        for (int i = 0; i < NQ; ++i) s += a[i] * Wx[(DTRANK + NQ + j) * NQ + i];
        Cbuf[(size_t)idx * NQ + j] = s;
    }
#pragma unroll
    for (int q = 0; q < NQ; ++q) {
        float s = bdt[q];
#pragma unroll
        for (int j = 0; j < DTRANK; ++j) s += dtr[j] * Wdt[q * DTRANK + j];
        float sp = (s > 20.f) ? s : log1pf(expf(s));   // softplus
        dtang[(size_t)idx * NQ + q] = tanhf(sp) * PI_F;
    }
}

// =====================================================================
// Kernel B: build fixed structures on device:
//   perm[64], Wph[64][6]  (QSVT = perm^4 * diag(exp(i*Wph@dt)))
//   U = 64x64 complex unitary of the fixed-parameter ansatz
// =====================================================================
__device__ __forceinline__ void qm_apply1q(float (*Ur)[DIM], float (*Ui)[DIM], int q,
                                           float m00r, float m00i, float m01r, float m01i,
                                           float m10r, float m10i, float m11r, float m11i,
                                           int tid) {
    int p = 5 - q;
    for (int it = tid; it < DIM * 32; it += 256) {
        int col = it & 63;
        int pr  = it >> 6;                       // 0..31 pair id
        int i0  = ((pr >> p) << (p + 1)) | (pr & ((1 << p) - 1));
        int i1  = i0 | (1 << p);
        float ar = Ur[i0][col], ai = Ui[i0][col];
        float br = Ur[i1][col], bi = Ui[i1][col];
        Ur[i0][col] = m00r * ar - m00i * ai + m01r * br - m01i * bi;
        Ui[i0][col] = m00r * ai + m00i * ar + m01r * bi + m01i * br;
        Ur[i1][col] = m10r * ar - m10i * ai + m11r * br - m11i * bi;
        Ui[i1][col] = m10r * ai + m10i * ar + m11r * bi + m11i * br;
    }
    __syncthreads();
}

__device__ __forceinline__ void qm_applycrx(float (*Ur)[DIM], float (*Ui)[DIM],
                                            int c, int t, float theta, int tid) {
    int pc = 5 - c, pt = 5 - t;
    float ch = cosf(0.5f * theta), sh = sinf(0.5f * theta);
    for (int it = tid; it < DIM * 16; it += 256) {
        int col = it & 63;
        int j   = it >> 6;                       // 0..15: free bits
        int i0 = 0, b = 0;
#pragma unroll
        for (int pos = 0; pos < 6; ++pos) {
            if (pos == pc) i0 |= (1 << pos);
            else if (pos != pt) { i0 |= ((j >> b) & 1) << pos; ++b; }
        }
        int i1 = i0 | (1 << pt);
        float ar = Ur[i0][col], ai = Ui[i0][col];
        float br = Ur[i1][col], bi = Ui[i1][col];
        // RX = [[ch, -i sh],[-i sh, ch]]
        Ur[i0][col] =  ch * ar + sh * bi;
        Ui[i0][col] =  ch * ai - sh * br;
        Ur[i1][col] =  sh * ai + ch * br;
        Ui[i1][col] = -sh * ar + ch * bi;
    }
    __syncthreads();
}

__global__ void qm_setup_kernel(const float* __restrict__ poly,
                                const float* __restrict__ qsvt,
                                const float* __restrict__ cp,
                                float* __restrict__ UrG, float* __restrict__ UiG,
                                float* __restrict__ WphG, int* __restrict__ permG) {
    __shared__ float sUr[DIM][DIM];
    __shared__ float sUi[DIM][DIM];
    int tid = threadIdx.x;

    // ---- part 1: QSVT permutation + phase-weight matrix (no barriers) ----
    if (tid < DIM) {
        int cur = tid;
        float w[NQ] = {0, 0, 0, 0, 0, 0};
        for (int deg = 0; deg <= 3; ++deg) {
            float coeff = poly[deg];
#pragma unroll
            for (int i = 0; i < NQ; ++i) {
                float k = coeff * PI_F * (deg < 3 ? qsvt[deg * NQ + i] : 1.0f);
                int bit = (cur >> (5 - i)) & 1;
                // amp *= exp(-i*th/2) for bit0, exp(+i*th/2) for bit1; th = k*dt_i
                w[i] += (bit ? 0.5f : -0.5f) * k;
            }
            // CNOT ladder: fwd chain, wrap, reverse chain (pure index perm)
            for (int i = 0; i < 5; ++i)
                if ((cur >> (5 - i)) & 1) cur ^= (1 << (5 - (i + 1)));
            if (cur & 1) cur ^= (1 << 5);                       // CNOT(5,0)
            for (int i = 5; i >= 1; --i)
                if ((cur >> (5 - i)) & 1) cur ^= (1 << (5 - (i - 1)));
        }
        permG[tid] = cur;
#pragma unroll
        for (int i = 0; i < NQ; ++i) WphG[tid * NQ + i] = w[i];
    }

    // ---- part 2: build U (identity, then gate-by-gate, left-multiplied) ----
    for (int k = tid; k < DIM * DIM; k += 256) {
        sUr[k >> 6][k & 63] = ((k >> 6) == (k & 63)) ? 1.0f : 0.0f;
        sUi[k >> 6][k & 63] = 0.0f;
    }
    __syncthreads();

    int p = 0;
    for (int layer = 0; layer < 2; ++layer) {
        for (int q = 0; q < NQ; ++q) {
            float t0 = cp[p], t1 = cp[p + 1], t2 = cp[p + 2];
            p += 3;
            float c, s;
            c = cosf(0.5f * t0); s = sinf(0.5f * t0);           // RX
            qm_apply1q(sUr, sUi, q, c, 0, 0, -s, 0, -s, c, 0, tid);
            c = cosf(0.5f * t1); s = sinf(0.5f * t1);           // RY
            qm_apply1q(sUr, sUi, q, c, 0, -s, 0, s, 0, c, 0, tid);
            c = cosf(0.5f * t2); s = sinf(0.5f * t2);           // RZ
            qm_apply1q(sUr, sUi, q, c, -s, 0, 0, 0, 0, c, s, tid);
        }
        for (int i = 0; i < NQ; ++i) qm_applycrx(sUr, sUi, i, (i + 1) % NQ, cp[p++], tid);
        for (int i = NQ - 1; i >= 0; --i) qm_applycrx(sUr, sUi, i, (i + 5) % NQ, cp[p++], tid);
    }
    for (int k = tid; k < DIM * DIM; k += 256) {
        UrG[k] = sUr[k >> 6][k & 63];
        UiG[k] = sUi[k >> 6][k & 63];
    }
}

// =====================================================================
// Kernel C: the 512-step sequential scan. 16 blocks x 256 threads (8 waves).
// Block owns a 16-wide batch tile; U lives in registers (A-fragments) and
// LDS; state panels live in LDS for all 512 steps.
// Per step: complex 64x64 x 64x16 GEMM via v_wmma_f32_16x16x4_f32 with two
// independent accumulator chains; final RY gates folded into the observable
// rotation (Heisenberg picture), so no post-GEMM state update is needed.
// =====================================================================
__global__ void __launch_bounds__(256)
qm_scan_kernel(const float* __restrict__ angG,
               const float* __restrict__ dtangG,
               const float* __restrict__ CbufG,
               const float* __restrict__ UrG, const float* __restrict__ UiG,
               const float* __restrict__ WphG, const int* __restrict__ permG,
               const float* __restrict__ WcG, const float* __restrict__ DvG,
               float* __restrict__ out) {
    __shared__ float lUr[DIM][DIM];        // 16 KB
    __shared__ float lUi[DIM][DIM];        // 16 KB
    __shared__ float Sr[DIM][BT], Si[DIM][BT], Sn[DIM][BT];   // Sn = -Si
    __shared__ float Dr[DIM][BT], Di[DIM][BT];
    __shared__ float Wl[DIM][NQ];
    __shared__ int   pl[DIM];
    __shared__ float hc[BT][NQ], hs[BT][NQ], dtv[BT][NQ], xv[BT][NQ], Cv[BT][NQ];
    __shared__ float meas[BT][OUTD];
    __shared__ float WcL[OUTD * NQ], DvL[OUTD];
    __shared__ float hcur[BT][NQ];

    const int tid  = threadIdx.x;
    const int b0   = blockIdx.x * BT;

    for (int k = tid; k < DIM * DIM; k += 256) {
        lUr[k >> 6][k & 63] = UrG[k];
        lUi[k >> 6][k & 63] = UiG[k];
    }
    for (int k = tid; k < DIM * NQ; k += 256) Wl[k / NQ][k % NQ] = WphG[k];
    if (tid < DIM) pl[tid] = permG[tid];
    if (tid < OUTD * NQ) WcL[tid] = WcG[tid];
    if (tid < OUTD) DvL[tid] = DvG[tid];
    if (tid < BT * NQ) hcur[tid / NQ][tid % NQ] = 0.0f;   // h0 = 0
    __syncthreads();

    // ---- wave-static WMMA geometry; hoist A (U) fragments to registers ----
    const int wave = tid >> 5, lane = tid & 31;
    const int comp = wave & 1;             // 0: real part, 1: imag part
    const int rt   = wave >> 1;            // row tile 0..3
    const int hi   = lane >> 4;            // K-half select per ISA layout
    const int am   = rt * 16 + (lane & 15);
    const int ncol = lane & 15;
    v2f aR[16], aI[16];
#pragma unroll
    for (int kc = 0; kc < 16; ++kc) {
        int ka = kc * 4 + hi * 2;
        aR[kc] = v2f{ lUr[am][ka], lUr[am][ka + 1] };
        aI[kc] = v2f{ lUi[am][ka], lUi[am][ka + 1] };
    }

    for (int t = 0; t < SEQ; ++t) {
        // ---- stage per-step inputs + carried h trig ----
        if (tid < BT * NQ) {
            int col = tid / NQ, q = tid % NQ;
            size_t base = ((size_t)(b0 + col) * SEQ + t) * NQ + q;
            float dt = dtangG[base];
            dtv[col][q] = dt;
            xv[col][q]  = angG[base];
            Cv[col][q]  = CbufG[base];
            float h = 0.5f * hcur[col][q];
            hc[col][q] = cosf(h);
            hs[col][q] = sinf(h);
        }
        if (t + 1 < SEQ && tid < BT) {       // hint next step's lines into cache
            size_t nb = ((size_t)(b0 + tid) * SEQ + (t + 1)) * NQ;
            __builtin_prefetch(&dtangG[nb], 0, 1);
            __builtin_prefetch(&angG[nb], 0, 1);
            __builtin_prefetch(&CbufG[nb], 0, 1);
        }
        __syncthreads();

        // ---- build post-QSVT state: S[perm[b]] = psi(b) * exp(i*Wph@dt) ----
        for (int it = tid; it < DIM * BT; it += 256) {
            int col = it & (BT - 1);
            int bix = it >> 4;               // basis index 0..63
            float psi = 1.0f, ph = 0.0f;
#pragma unroll
            for (int q = 0; q < NQ; ++q) {
                int bit = (bix >> (5 - q)) & 1;
                psi *= bit ? hs[col][q] : hc[col][q];
                ph  += Wl[bix][q] * dtv[col][q];
            }
            float cr = psi * cosf(ph), ci = psi * sinf(ph);
            int dst = pl[bix];
            Sr[dst][col] = cr;
            Si[dst][col] = ci;
            Sn[dst][col] = -ci;
        }
        __syncthreads();

        // ---- D = U * S (complex) via WMMA f32 16x16x4, all 8 waves ----
        {
            float (*B1)[BT] = comp ? Si : Sr;    // Ur * B1
            float (*B2)[BT] = comp ? Sr : Sn;    // Ui * B2  (Sn = -Si)
            v8f acc1 = {}, acc2 = {};            // two independent chains
#pragma unroll
            for (int kc = 0; kc < 16; ++kc) {
                int ka = kc * 4 + hi * 2;
                v2f b1 = v2f{ B1[ka][ncol], B1[ka + 1][ncol] };
                acc1 = __builtin_amdgcn_wmma_f32_16x16x4_f32(
                        false, aR[kc], false, b1, (short)0, acc1, false, false);
                v2f b2 = v2f{ B2[ka][ncol], B2[ka + 1][ncol] };
                acc2 = __builtin_amdgcn_wmma_f32_16x16x4_f32(
                        false, aI[kc], false, b2, (short)0, acc2, false, false);
            }
            float (*Dp)[BT] = comp ? Di : Dr;
#pragma unroll
            for (int r = 0; r < 8; ++r)
                Dp[rt * 16 + r + hi * 8][ncol] = acc1[r] + acc2[r];
        }
        __syncthreads();

        // ---- expectations <X>,<Y>,<Z>; final RY gates folded in via
        //      Heisenberg rotation: X'=cX+sZ, Y'=Y, Z'=cZ-sX (th=x*dt) ----
        if (tid < BT * NQ) {
            int col = tid / NQ, q = tid % NQ;
            int p = 5 - q;
            float zr = 0.f, zi = 0.f, zd = 0.f;
            for (int pr = 0; pr < 32; ++pr) {
                int i0 = ((pr >> p) << (p + 1)) | (pr & ((1 << p) - 1));
                int i1 = i0 | (1 << p);
                float ar = Dr[i0][col], ai = Di[i0][col];
                float br = Dr[i1][col], bi = Di[i1][col];
                zr += ar * br + ai * bi;                  // Re(conj(a)*b)
                zi += ar * bi - ai * br;                  // Im(conj(a)*b)
                zd += (ar * ar + ai * ai) - (br * br + bi * bi);
            }
            float X = 2.0f * zr, Y = 2.0f * zi, Z = zd;
            float th = xv[col][q] * dtv[col][q];
            float c = cosf(th), s = sinf(th);
            float zf = c * Z - s * X;
            meas[col][q]          = c * X + s * Z;
            meas[col][NQ + q]     = Y;
            meas[col][2 * NQ + q] = zf;
            hcur[col][q] = zf;
        }
        __syncthreads();

        // ---- epilogue: out = (C@Wc.T)*meas + D*tile(angles,3) ----
        for (int it = tid; it < BT * OUTD; it += 256) {
            int col = it / OUTD, o = it % OUTD;
            float acc = 0.f;
#pragma unroll
            for (int q = 0; q < NQ; ++q) acc += Cv[col][q] * WcL[o * NQ + q];
            out[((size_t)(b0 + col) * SEQ + t) * OUTD + o] =
                acc * meas[col][o] + DvL[o] * xv[col][o % NQ];
        }
        __syncthreads();
    }
}

// =====================================================================
// Launch
// =====================================================================
extern "C" void kernel_launch(void* const* d_in, const int* in_sizes, int n_in,
                              void* d_out, int out_size, void* d_ws, size_t ws_size,
                              hipStream_t stream) {
    (void)in_sizes; (void)n_in; (void)out_size; (void)ws_size;
    const float* angles  = (const float*)d_in[0];
    const float* Wx      = (const float*)d_in[1];
    const float* Wdt     = (const float*)d_in[2];
    const float* bdt     = (const float*)d_in[3];
    const float* poly    = (const float*)d_in[4];
    const float* qsvt    = (const float*)d_in[5];
    const float* cparams = (const float*)d_in[6];
    const float* Dvec    = (const float*)d_in[7];
    const float* Wc      = (const float*)d_in[8];
    float* out = (float*)d_out;

    // workspace layout (floats): dtang | Cbuf | Ur | Ui | Wph | perm(int)
    float* ws    = (float*)d_ws;
    float* dtang = ws;                                  // BATCH*SEQ*NQ
    float* Cbuf  = dtang + (size_t)BATCH * SEQ * NQ;    // BATCH*SEQ*NQ
    float* Ur    = Cbuf + (size_t)BATCH * SEQ * NQ;     // 4096
    float* Ui    = Ur + DIM * DIM;                      // 4096
    float* Wph   = Ui + DIM * DIM;                      // 384
    int*   perm  = (int*)(Wph + DIM * NQ);              // 64

    qm_precompute_kernel<<<(BATCH * SEQ + 255) / 256, 256, 0, stream>>>(
        angles, Wx, Wdt, bdt, dtang, Cbuf);
    qm_setup_kernel<<<1, 256, 0, stream>>>(poly, qsvt, cparams, Ur, Ui, Wph, perm);
    qm_scan_kernel<<<NBLK, 256, 0, stream>>>(
        angles, dtang, Cbuf, Ur, Ui, Wph, perm, Wc, Dvec, out);
}